// behavior_model_45818711114225
// MI455X (gfx1250) — compile-verified
//
#include <hip/hip_runtime.h>
#include <hip/hip_bf16.h>

// ---------------------------------------------------------------------------
// MI455X (gfx1250) implementation.
// Pipeline: roi_pool -> GEMM1 (wmma) -> GEMM2 (wmma, scatter epilogue)
//           -> FC1 (wmma, bias+relu) -> FC2 (small, wave dot products)
// GEMMs: fp32 V_WMMA_F32_16X16X4_F32, LDS double-buffered. Staging uses
// GLOBAL_LOAD_ASYNC_TO_LDS_B128 (ASYNCcnt) when the toolchain exposes the
// builtin, else coalesced b128 loads + ds_store_b128.
// ---------------------------------------------------------------------------

typedef __attribute__((ext_vector_type(2))) float v2f;
typedef __attribute__((ext_vector_type(4))) float v4f;
typedef __attribute__((ext_vector_type(8))) float v8f;
typedef int b128_t __attribute__((vector_size(16)));   // matches builtin param

#define R_TOTAL   1024      // 64 batches * 16 rois
#define C_IN      1024
#define FMAP_HW   14
#define POS       9         // 3x3 pooled positions
#define M_GEMM    (R_TOTAL * POS)   // 9216

#define KC        32                 // K-chunk staged in LDS
#define ASTRIDE   (KC + 4)           // padded row stride (floats): 144B rows

// ---- gfx1250 async global->LDS path (guarded) -----------------------------
#if defined(__has_builtin)
#if __has_builtin(__builtin_amdgcn_global_load_async_to_lds_b128)
#define HAVE_ASYNC_LDS 1
#endif
#endif

#ifdef HAVE_ASYNC_LDS
// builtin prototype (from hipcc diagnostic): param0 = v4i AS1*, param1 = AS3*,
// then imm offset, imm cpol. Address-space conversion via integer round-trip
// (low 32 bits of a generic LDS address are the LDS byte offset).
#define ASYNC_CP_B128(gp, lp)                                                  \
    __builtin_amdgcn_global_load_async_to_lds_b128(                            \
        (__attribute__((address_space(1))) b128_t*)(unsigned long long)(gp),   \
        (__attribute__((address_space(3))) b128_t*)(unsigned int)              \
            (unsigned long long)(lp),                                          \
        0, 0)
#if __has_builtin(__builtin_amdgcn_s_wait_asynccnt)
#define WAIT_ASYNC(n) __builtin_amdgcn_s_wait_asynccnt(n)
#else
#define WAIT_ASYNC(n) asm volatile("s_wait_asynccnt %0" ::"i"(n) : "memory")
#endif
#endif

// ---------------------------------------------------------------------------
// ROI align: one block per ROI (grid.x = 1024), 256 threads across channels
// (each thread handles 4 channels). Output layout: pooled[(r*9 + p)*1024 + c]
// i.e. A-matrix of GEMM1, rows = (roi, position), cols = channel.
// ---------------------------------------------------------------------------
__global__ __launch_bounds__(256)
void roi_pool_kernel(const float* __restrict__ fmap,
                     const float* __restrict__ boxes,
                     float* __restrict__ pooled)
{
    const int r = blockIdx.x;
    const int b = r >> 4;                 // batch_idx = repeat(arange(64),16)
    const float* bx = boxes + r * 4;

    float x1 = fminf(fmaxf(bx[0], 0.001f), 13.999f);
    float y1 = fminf(fmaxf(bx[1], 0.001f), 13.999f);
    float x2 = fminf(fmaxf(bx[2], 0.001f), 13.999f);
    float y2 = fminf(fmaxf(bx[3], 0.001f), 13.999f);

    const float roi_w = fmaxf(x2 - x1, 1.0f);
    const float roi_h = fmaxf(y2 - y1, 1.0f);
    const float bin_w = roi_w * (1.0f / 3.0f);
    const float bin_h = roi_h * (1.0f / 3.0f);

    int   ylo[6], xlo[6];
    float fly[6], fhy[6], flx[6], fhx[6];
#pragma unroll
    for (int i = 0; i < 6; ++i) {
        const float g = (i + 0.5f) * 0.5f;

        float yy = y1 + bin_h * g;
        bool oky = (yy >= -1.0f) && (yy <= (float)FMAP_HW);
        float yc = fminf(fmaxf(yy, 0.0f), (float)(FMAP_HW - 1));
        int yl = (int)floorf(yc); yl = yl > (FMAP_HW - 2) ? (FMAP_HW - 2) : yl;
        float ly = yc - (float)yl;
        float hy = 1.0f - ly;
        if (!oky) { ly = 0.0f; hy = 0.0f; }
        ylo[i] = yl; fly[i] = ly; fhy[i] = hy;

        float xx = x1 + bin_w * g;
        bool okx = (xx >= -1.0f) && (xx <= (float)FMAP_HW);
        float xc = fminf(fmaxf(xx, 0.0f), (float)(FMAP_HW - 1));
        int xl = (int)floorf(xc); xl = xl > (FMAP_HW - 2) ? (FMAP_HW - 2) : xl;
        float lx = xc - (float)xl;
        float hx = 1.0f - lx;
        if (!okx) { lx = 0.0f; hx = 0.0f; }
        xlo[i] = xl; flx[i] = lx; fhx[i] = hx;
    }

#pragma unroll
    for (int cc = 0; cc < 4; ++cc) {
        const int c = threadIdx.x + cc * 256;
        const float* fb = fmap + ((size_t)(b * C_IN + c)) * (FMAP_HW * FMAP_HW);
#pragma unroll
        for (int p = 0; p < POS; ++p) {
            const int ph = p / 3, pw = p - 3 * (p / 3);
            float acc = 0.0f;
#pragma unroll
            for (int sy = 0; sy < 2; ++sy) {
#pragma unroll
                for (int sx = 0; sx < 2; ++sx) {
                    const int gy = ph * 2 + sy;
                    const int gx = pw * 2 + sx;
                    const float* base = fb + ylo[gy] * FMAP_HW + xlo[gx];
                    const float v00 = base[0];
                    const float v01 = base[1];
                    const float v10 = base[FMAP_HW];
                    const float v11 = base[FMAP_HW + 1];
                    acc += fhy[gy] * (fhx[gx] * v00 + flx[gx] * v01)
                         + fly[gy] * (fhx[gx] * v10 + flx[gx] * v11);
                }
            }
            pooled[((size_t)r * POS + p) * C_IN + c] = acc * 0.25f;
        }
    }
}

// ---------------------------------------------------------------------------
// fp32 WMMA GEMM:   D[M,N] = A[M,K] * B[N,K]^T   (both operands K-major rows).
//
// Block: 256 threads = 8 waves, 4 (M) x 2 (N). Wave tile 16M x 64N, block
// tile 64M x 128N. K chunked at 32, double-buffered through LDS.
//
// Async path (gfx1250): per chunk, each thread issues 6 b128 async
// global->LDS copies (256 thr * 16 B * 6 = 24 KB chunk); pipeline is
//   barrier -> issue chunk k+1 -> s_wait_asynccnt 6 (chunk k complete,
//   in-order) -> barrier -> 8x4 WMMA burst from LDS buffer k&1.
// Fallback path: b128 global loads -> regs -> ds_store_b128.
//
// LDS rows padded to 36 floats (144 B): 16B-aligned b128 stores and
// bank-conflict-free ds_load_b64 fragment reads.
//
// EPI: 0 = plain row-major store, 1 = conv2 scatter into fc1 layout
//      (out[r*2304 + n*9 + p], m = r*9+p), 2 = bias + relu.
// ---------------------------------------------------------------------------
template <int EPI>
__global__ __launch_bounds__(256)
void gemm_wmma_f32(const float* __restrict__ A,
                   const float* __restrict__ B,
                   float* __restrict__ D,
                   int M, int N, int K,
                   const float* __restrict__ bias)
{
    __shared__ float as[2][64  * ASTRIDE];   // 18.4 KB
    __shared__ float bs[2][128 * ASTRIDE];   // 36.9 KB

    const int lane = threadIdx.x & 31;
    const int wave = threadIdx.x >> 5;
    const int half = lane >> 4;          // 0 or 1
    const int l15  = lane & 15;

    const int wm = wave & 3;             // 0..3  (M sub-tile)
    const int wn = wave >> 2;            // 0..1  (N sub-tile)

    // fragment read offsets (floats) inside an LDS buffer
    const int aoff  = (wm * 16 + l15) * ASTRIDE + 2 * half;
    const int boff0 = (wn * 64 +  0 + l15) * ASTRIDE + 2 * half;
    const int boff1 = (wn * 64 + 16 + l15) * ASTRIDE + 2 * half;
    const int boff2 = (wn * 64 + 32 + l15) * ASTRIDE + 2 * half;
    const int boff3 = (wn * 64 + 48 + l15) * ASTRIDE + 2 * half;

    const int NC = K / KC;

    v8f acc0 = {}, acc1 = {}, acc2 = {}, acc3 = {};

#ifdef HAVE_ASYNC_LDS
    // ---- async copy mapping: thread covers one 16B slice in each of 6 row
    // groups (A: 2 groups of 32 rows, B: 4 groups of 32 rows) ---------------
    const int car = threadIdx.x >> 3;        // 0..31 row within group
    const int cac = (threadIdx.x & 7) * 4;   // col: 0,4,...,28

    const float* ag0 = A + (size_t)(blockIdx.y * 64       + car) * K + cac;
    const float* ag1 = A + (size_t)(blockIdx.y * 64  + 32 + car) * K + cac;
    const float* bg0 = B + (size_t)(blockIdx.x * 128      + car) * K + cac;
    const float* bg1 = B + (size_t)(blockIdx.x * 128 + 32 + car) * K + cac;
    const float* bg2 = B + (size_t)(blockIdx.x * 128 + 64 + car) * K + cac;
    const float* bg3 = B + (size_t)(blockIdx.x * 128 + 96 + car) * K + cac;

    const int al0 = (car      ) * ASTRIDE + cac;
    const int al1 = (car + 32 ) * ASTRIDE + cac;
    const int bl0 = (car      ) * ASTRIDE + cac;
    const int bl1 = (car + 32 ) * ASTRIDE + cac;
    const int bl2 = (car + 64 ) * ASTRIDE + cac;
    const int bl3 = (car + 96 ) * ASTRIDE + cac;

    // prologue: chunk 0 -> buffer 0
    ASYNC_CP_B128(ag0, &as[0][al0]);
    ASYNC_CP_B128(ag1, &as[0][al1]);
    ASYNC_CP_B128(bg0, &bs[0][bl0]);
    ASYNC_CP_B128(bg1, &bs[0][bl1]);
    ASYNC_CP_B128(bg2, &bs[0][bl2]);
    ASYNC_CP_B128(bg3, &bs[0][bl3]);

    for (int kc = 0; kc < NC; ++kc) {
        const int buf = kc & 1;
        __syncthreads();                 // all reads of LDS buffer buf^1 retired
        if (kc + 1 < NC) {
            const int ko = (kc + 1) * KC;
            const int nb = buf ^ 1;
            ASYNC_CP_B128(ag0 + ko, &as[nb][al0]);
            ASYNC_CP_B128(ag1 + ko, &as[nb][al1]);
            ASYNC_CP_B128(bg0 + ko, &bs[nb][bl0]);
            ASYNC_CP_B128(bg1 + ko, &bs[nb][bl1]);
            ASYNC_CP_B128(bg2 + ko, &bs[nb][bl2]);
            ASYNC_CP_B128(bg3 + ko, &bs[nb][bl3]);
            WAIT_ASYNC(6);               // chunk kc done (in-order), kc+1 in flight
        } else {
            WAIT_ASYNC(0);
        }
        __syncthreads();                 // chunk kc visible block-wide

#pragma unroll
        for (int ks = 0; ks < 8; ++ks) {
            const int k = ks * 4;
            const v2f a  = *(const v2f*)&as[buf][aoff  + k];
            const v2f b0 = *(const v2f*)&bs[buf][boff0 + k];
            const v2f b1 = *(const v2f*)&bs[buf][boff1 + k];
            const v2f b2 = *(const v2f*)&bs[buf][boff2 + k];
            const v2f b3 = *(const v2f*)&bs[buf][boff3 + k];
            acc0 = __builtin_amdgcn_wmma_f32_16x16x4_f32(false, a, false, b0,
                                                         (short)0, acc0, false, false);
            acc1 = __builtin_amdgcn_wmma_f32_16x16x4_f32(false, a, false, b1,
                                                         (short)0, acc1, false, false);
            acc2 = __builtin_amdgcn_wmma_f32_16x16x4_f32(false, a, false, b2,
                                                         (short)0, acc2, false, false);
            acc3 = __builtin_amdgcn_wmma_f32_16x16x4_f32(false, a, false, b3,
                                                         (short)0, acc3, false, false);
        }
    }
#else
    // ---- fallback: b128 global loads -> regs -> ds_store_b128 -------------
    const int ldr = threadIdx.x >> 2;        // 0..63
    const int ldc = (threadIdx.x & 3) * 8;   // 0,8,16,24
    const float* agp  = A + (size_t)(blockIdx.y * 64        + ldr) * K + ldc;
    const float* bgp0 = B + (size_t)(blockIdx.x * 128       + ldr) * K + ldc;
    const float* bgp1 = B + (size_t)(blockIdx.x * 128 + 64  + ldr) * K + ldc;

    const int sta = ldr * ASTRIDE + ldc;
    const int stb = (64 + ldr) * ASTRIDE + ldc;

    v4f ra0, ra1, rb0, rb1, rb2, rb3;
    ra0 = *(const v4f*)(agp);       ra1 = *(const v4f*)(agp + 4);
    rb0 = *(const v4f*)(bgp0);      rb1 = *(const v4f*)(bgp0 + 4);
    rb2 = *(const v4f*)(bgp1);      rb3 = *(const v4f*)(bgp1 + 4);
    *(v4f*)&as[0][sta]     = ra0;   *(v4f*)&as[0][sta + 4] = ra1;
    *(v4f*)&bs[0][sta]     = rb0;   *(v4f*)&bs[0][sta + 4] = rb1;
    *(v4f*)&bs[0][stb]     = rb2;   *(v4f*)&bs[0][stb + 4] = rb3;
    __syncthreads();

    for (int kc = 0; kc < NC; ++kc) {
        const int buf = kc & 1;
        if (kc + 1 < NC) {
            const int ko = (kc + 1) * KC;
            ra0 = *(const v4f*)(agp  + ko);  ra1 = *(const v4f*)(agp  + ko + 4);
            rb0 = *(const v4f*)(bgp0 + ko);  rb1 = *(const v4f*)(bgp0 + ko + 4);
            rb2 = *(const v4f*)(bgp1 + ko);  rb3 = *(const v4f*)(bgp1 + ko + 4);
        }
#pragma unroll
        for (int ks = 0; ks < 8; ++ks) {
            const int k = ks * 4;
            const v2f a  = *(const v2f*)&as[buf][aoff  + k];
            const v2f b0 = *(const v2f*)&bs[buf][boff0 + k];
            const v2f b1 = *(const v2f*)&bs[buf][boff1 + k];
            const v2f b2 = *(const v2f*)&bs[buf][boff2 + k];
            const v2f b3 = *(const v2f*)&bs[buf][boff3 + k];
            acc0 = __builtin_amdgcn_wmma_f32_16x16x4_f32(false, a, false, b0,
                                                         (short)0, acc0, false, false);
            acc1 = __builtin_amdgcn_wmma_f32_16x16x4_f32(false, a, false, b1,
                                                         (short)0, acc1, false, false);
            acc2 = __builtin_amdgcn_wmma_f32_16x16x4_f32(false, a, false, b2,
                                                         (short)0, acc2, false, false);
            acc3 = __builtin_amdgcn_wmma_f32_16x16x4_f32(false, a, false, b3,
                                                         (short)0, acc3, false, false);
        }
        if (kc + 1 < NC) {
            const int nb = buf ^ 1;
            __syncthreads();
            *(v4f*)&as[nb][sta]     = ra0;  *(v4f*)&as[nb][sta + 4] = ra1;
            *(v4f*)&bs[nb][sta]     = rb0;  *(v4f*)&bs[nb][sta + 4] = rb1;
            *(v4f*)&bs[nb][stb]     = rb2;  *(v4f*)&bs[nb][stb + 4] = rb3;
            __syncthreads();
        }
    }
#endif

    const int mbase = blockIdx.y * 64 + wm * 16;
    const int nbase = blockIdx.x * 128 + wn * 64;

    v8f accs[4] = {acc0, acc1, acc2, acc3};
#pragma unroll
    for (int t = 0; t < 4; ++t) {
        const int n = nbase + t * 16 + l15;
#pragma unroll
        for (int i = 0; i < 8; ++i) {
            const int m = mbase + i + 8 * half;
            float v = accs[t][i];
            if (EPI == 1) {
                const int r = m / POS;
                const int p = m - r * POS;
                D[(size_t)r * 2304 + n * POS + p] = v;
            } else if (EPI == 2) {
                v += bias[n];
                v = v > 0.0f ? v : 0.0f;
                D[(size_t)m * N + n] = v;
            } else {
                D[(size_t)m * N + n] = v;
            }
        }
    }
}

// ---------------------------------------------------------------------------
// FC2: logits[r, j] = sum_k h[r,k] * w2[j,k] + b2[j],  N=27. One wave per row,
// lane j (< 27) owns one output. 1024 rows -> 128 blocks of 8 waves.
// ---------------------------------------------------------------------------
__global__ __launch_bounds__(256)
void fc2_kernel(const float* __restrict__ h,
                const float* __restrict__ w2,
                const float* __restrict__ b2,
                float* __restrict__ out)
{
    const int row  = (blockIdx.x * blockDim.x + threadIdx.x) >> 5;
    const int lane = threadIdx.x & 31;
    if (lane >= 27) return;

    const float* hr = h  + (size_t)row * 1024;
    const float* wr = w2 + (size_t)lane * 1024;
    float sum = 0.0f;
#pragma unroll 4
    for (int k = 0; k < 1024; k += 4) {
        sum += hr[k + 0] * wr[k + 0];
        sum += hr[k + 1] * wr[k + 1];
        sum += hr[k + 2] * wr[k + 2];
        sum += hr[k + 3] * wr[k + 3];
    }
    out[(size_t)row * 27 + lane] = sum + b2[lane];
}

// ---------------------------------------------------------------------------
extern "C" void kernel_launch(void* const* d_in, const int* in_sizes, int n_in,
                              void* d_out, int out_size, void* d_ws, size_t ws_size,
                              hipStream_t stream)
{
    const float* fmap    = (const float*)d_in[0];  // [64,1024,14,14]
    const float* boxes   = (const float*)d_in[1];  // [64,16,4]
    const float* w_conv1 = (const float*)d_in[2];  // [512,1024]
    const float* w_conv2 = (const float*)d_in[3];  // [256,512]
    const float* w_fc1   = (const float*)d_in[4];  // [1024,2304]
    const float* b_fc1   = (const float*)d_in[5];  // [1024]
    const float* w_fc2   = (const float*)d_in[6];  // [27,1024]
    const float* b_fc2   = (const float*)d_in[7];  // [27]
    float* logits        = (float*)d_out;          // [1024,27]

    // Workspace layout (fp32):
    float* pooled = (float*)d_ws;                       // [9216,1024]
    float* h1     = pooled + (size_t)M_GEMM * 1024;     // [9216,512]
    float* xfc    = h1     + (size_t)M_GEMM * 512;      // [1024,2304]
    float* hfc    = xfc    + (size_t)R_TOTAL * 2304;    // [1024,1024]

    // 1. ROI align -> GEMM-ready activations
    roi_pool_kernel<<<R_TOTAL, 256, 0, stream>>>(fmap, boxes, pooled);

    // 2. conv1: [9216,1024] x [512,1024]^T -> h1 [9216,512]
    gemm_wmma_f32<0><<<dim3(512 / 128, M_GEMM / 64), 256, 0, stream>>>(
        pooled, w_conv1, h1, M_GEMM, 512, 1024, nullptr);

    // 3. conv2: [9216,512] x [256,512]^T -> scattered into xfc [1024,2304]
    gemm_wmma_f32<1><<<dim3(256 / 128, M_GEMM / 64), 256, 0, stream>>>(
        h1, w_conv2, xfc, M_GEMM, 256, 512, nullptr);

    // 4. fc1: [1024,2304] x [1024,2304]^T + bias, relu -> hfc [1024,1024]
    gemm_wmma_f32<2><<<dim3(1024 / 128, R_TOTAL / 64), 256, 0, stream>>>(
        xfc, w_fc1, hfc, R_TOTAL, 1024, 2304, b_fc1);

    // 5. fc2: [1024,1024] x [27,1024]^T + bias -> logits [1024,27]
    fc2_kernel<<<(R_TOTAL * 32) / 256, 256, 0, stream>>>(hfc, w_fc2, b_fc2, logits);
}